// GCN_15058155340594
// MI455X (gfx1250) — compile-verified
//
#include <hip/hip_runtime.h>
#include <hip/hip_bf16.h>

#define N_NODES 50000
#define N_EDGES 800000
#define N_GRAPHS 512
#define M_TILES 3125   // 50000 / 16, exact

typedef __attribute__((ext_vector_type(2))) float v2f;
typedef __attribute__((ext_vector_type(8))) float v8f;

// ---------------------------------------------------------------- utilities
__global__ void zero_f32(float* __restrict__ p, int n) {
    int i = blockIdx.x * blockDim.x + threadIdx.x;
    if (i < n) p[i] = 0.0f;
}

// Pad x [N,7] -> xp [N,8] with zero in column 7 (makes layer-1 K regular).
__global__ void pad_x(const float* __restrict__ x, float* __restrict__ xp) {
    int i = blockIdx.x * blockDim.x + threadIdx.x;
    if (i >= N_NODES * 8) return;
    int r = i >> 3, c = i & 7;
    xp[i] = (c < 7) ? x[r * 7 + c] : 0.0f;
}

// Pad W [7,16] -> Wp [8,16] with zero row 7 (flat index identical, 16 cols).
__global__ void pad_w(const float* __restrict__ W, float* __restrict__ Wp) {
    int i = threadIdx.x;  // 128 threads = 8*16
    if (i >= 128) return;
    Wp[i] = ((i >> 4) < 7) ? W[i] : 0.0f;
}

// ------------------------------------------------------- edge scatter (agg)
// agg[dst] += h[src], edge-parallel, 4 features per thread, f32 atomics.
// DIN here is always a multiple of 4 -> fully regular, no guards.
template <int DIN>
__global__ void edge_scatter(const float* __restrict__ h,
                             const int* __restrict__ src,
                             const int* __restrict__ dst,
                             float* __restrict__ agg) {
    constexpr int CH = DIN / 4;  // feature chunks of 4
    long t = (long)blockIdx.x * blockDim.x + threadIdx.x;
    int e = (int)(t / CH);
    int c = (int)(t % CH);
    if (e >= N_EDGES) return;
    int s = src[e];
    int d = dst[e];
    const float* hp = h + (long)s * DIN + c * 4;
    float* ap = agg + (long)d * DIN + c * 4;
#pragma unroll
    for (int j = 0; j < 4; ++j) atomicAdd(&ap[j], hp[j]);
}

// ------------------------------------------------ WMMA fused double GEMM
// hout = agg @ Wr + hin @ Wn + br    (all fp32, V_WMMA_F32_16X16X4_F32)
// One wave computes a 16 x (16*NT) C tile: NT accumulators share one A
// fragment per K-step (halves A traffic for NT=2). 4 waves / block, each on
// a distinct M tile (wave-uniform guard keeps EXEC all-1s for WMMA).
//
// fp32 WMMA fragment layouts (ISA 7.12.2):
//   A 16x4 : lane&15 = M row; half = lane>>4 selects K base {0}|{2};
//            vgpr0 = K+0, vgpr1 = K+1
//   B 4x16 : lane&15 = N col; same half/K mapping as A
//   C 16x16: vgpr j -> M = j + 8*half, N = lane&15
template <int DIN, int DOUT, int NT>
__global__ __launch_bounds__(128) void gemm_wmma(
    const float* __restrict__ agg, const float* __restrict__ Wr,
    const float* __restrict__ br, const float* __restrict__ hin,
    const float* __restrict__ Wn, float* __restrict__ hout) {
    static_assert(DIN % 4 == 0, "K must be regular");
    const int lane = threadIdx.x & 31;
    const int wave = threadIdx.x >> 5;
    const int mTile = blockIdx.x * 4 + wave;
    if (mTile >= M_TILES) return;
    const int nTile = blockIdx.y;

    const int half = lane >> 4;          // 0: K+{0,1}, 1: K+{2,3}
    const int arow = mTile * 16 + (lane & 15);
    const int col0 = nTile * (16 * NT) + (lane & 15);
    const int koff = half * 2;

    v8f c[NT];
#pragma unroll
    for (int t = 0; t < NT; ++t) c[t] = (v8f){};

    constexpr int KSTEPS = DIN / 4;

    // agg @ Wr
#pragma unroll
    for (int ks = 0; ks < KSTEPS; ++ks) {
        const int k = ks * 4 + koff;
        v2f a;
        a.x = agg[(long)arow * DIN + k + 0];
        a.y = agg[(long)arow * DIN + k + 1];
#pragma unroll
        for (int t = 0; t < NT; ++t) {
            v2f b;
            b.x = Wr[(long)(k + 0) * DOUT + col0 + 16 * t];
            b.y = Wr[(long)(k + 1) * DOUT + col0 + 16 * t];
            c[t] = __builtin_amdgcn_wmma_f32_16x16x4_f32(
                false, a, false, b, (short)0, c[t], false, false);
        }
    }
    // hin @ Wn
#pragma unroll
    for (int ks = 0; ks < KSTEPS; ++ks) {
        const int k = ks * 4 + koff;
        v2f a;
        a.x = hin[(long)arow * DIN + k + 0];
        a.y = hin[(long)arow * DIN + k + 1];
#pragma unroll
        for (int t = 0; t < NT; ++t) {
            v2f b;
            b.x = Wn[(long)(k + 0) * DOUT + col0 + 16 * t];
            b.y = Wn[(long)(k + 1) * DOUT + col0 + 16 * t];
            c[t] = __builtin_amdgcn_wmma_f32_16x16x4_f32(
                false, a, false, b, (short)0, c[t], false, false);
        }
    }

#pragma unroll
    for (int t = 0; t < NT; ++t) {
        const float bias = br[col0 + 16 * t];
#pragma unroll
        for (int j = 0; j < 8; ++j) {
            const int r = mTile * 16 + j + half * 8;
            hout[(long)r * DOUT + col0 + 16 * t] = c[t][j] + bias;
        }
    }
}

// ------------------------------------------------------------- global pool
__global__ void pool_kernel(const float* __restrict__ h,
                            const int* __restrict__ batch,
                            float* __restrict__ pooled) {
    long t = (long)blockIdx.x * blockDim.x + threadIdx.x;
    int node = (int)(t >> 6);  // 64 chunks of 4 floats per node (256 wide)
    int cch = (int)(t & 63);
    if (node >= N_NODES) return;
    int g = batch[node];
    const float* hp = h + (long)node * 256 + cch * 4;
    float* pp = pooled + (long)g * 256 + cch * 4;
#pragma unroll
    for (int j = 0; j < 4; ++j) atomicAdd(&pp[j], hp[j]);
}

// ------------------------------------------------------------ linear head
__global__ void final_linear(const float* __restrict__ pooled,
                             const float* __restrict__ Wlin,
                             const float* __restrict__ blin,
                             float* __restrict__ out) {
    int t = blockIdx.x * blockDim.x + threadIdx.x;
    if (t >= N_GRAPHS * 2) return;
    int g = t >> 1, c = t & 1;
    float acc = blin[c];
    const float* pp = pooled + (long)g * 256;
#pragma unroll 8
    for (int k = 0; k < 256; ++k) acc += pp[k] * Wlin[k * 2 + c];
    out[t] = acc;
}

// -------------------------------------------------------------- layer glue
template <int DIN, int DOUT, int NT>
static void run_layer(const float* hin, const float* Wr, const float* br,
                      const float* Wn, float* agg, float* hout,
                      const int* src, const int* dst, hipStream_t stream) {
    const int nAgg = N_NODES * DIN;
    zero_f32<<<(nAgg + 255) / 256, 256, 0, stream>>>(agg, nAgg);

    constexpr int CH = DIN / 4;
    const long nthr = (long)N_EDGES * CH;
    edge_scatter<DIN><<<(int)((nthr + 255) / 256), 256, 0, stream>>>(hin, src,
                                                                     dst, agg);

    dim3 grid((M_TILES + 3) / 4, DOUT / (16 * NT));
    gemm_wmma<DIN, DOUT, NT>
        <<<grid, 128, 0, stream>>>(agg, Wr, br, hin, Wn, hout);
}

extern "C" void kernel_launch(void* const* d_in, const int* in_sizes, int n_in,
                              void* d_out, int out_size, void* d_ws,
                              size_t ws_size, hipStream_t stream) {
    // setup_inputs() order:
    // 0:x 1:edge_index 2:batch 3..7:Wr[0..4] 8..12:br[0..4] 13..17:Wn[0..4]
    // 18:Wlin 19:blin
    const float* x = (const float*)d_in[0];
    const int* ei = (const int*)d_in[1];      // int32 (JAX default, x64 off)
    const int* batch = (const int*)d_in[2];
    const float* Wr[5], *br[5], *Wn[5];
    for (int i = 0; i < 5; ++i) {
        Wr[i] = (const float*)d_in[3 + i];
        br[i] = (const float*)d_in[8 + i];
        Wn[i] = (const float*)d_in[13 + i];
    }
    const float* Wlin = (const float*)d_in[18];
    const float* blin = (const float*)d_in[19];
    float* out = (float*)d_out;

    const int* src = ei;             // edge_index[0, :]
    const int* dst = ei + N_EDGES;   // edge_index[1, :]

    // workspace: bufA [50000,256] | bufB [50000,128] | agg [50000,128]
    //            | pooled [512,256] | xp [50000,8] | Wrp [8,16] | Wnp [8,16]
    //            (~105 MB total)
    float* bufA = (float*)d_ws;
    float* bufB = bufA + (size_t)N_NODES * 256;
    float* agg = bufB + (size_t)N_NODES * 128;
    float* pooled = agg + (size_t)N_NODES * 128;
    float* xp = pooled + (size_t)N_GRAPHS * 256;
    float* Wrp = xp + (size_t)N_NODES * 8;
    float* Wnp = Wrp + 128;

    // Regularize layer 1: K 7 -> 8 with zero padding (kills divergent guards)
    pad_x<<<(N_NODES * 8 + 255) / 256, 256, 0, stream>>>(x, xp);
    pad_w<<<1, 128, 0, stream>>>(Wr[0], Wrp);
    pad_w<<<1, 128, 0, stream>>>(Wn[0], Wnp);

    // 5 GraphConv layers, ping-pong so widths fit buffer capacities:
    // L1 out(16)->A, L2 out(32)->B, L3 out(64)->A, L4 out(128)->B,
    // L5 out(256)->A
    run_layer<8, 16, 1>(xp, Wrp, br[0], Wnp, agg, bufA, src, dst, stream);
    run_layer<16, 32, 2>(bufA, Wr[1], br[1], Wn[1], agg, bufB, src, dst,
                         stream);
    run_layer<32, 64, 2>(bufB, Wr[2], br[2], Wn[2], agg, bufA, src, dst,
                         stream);
    run_layer<64, 128, 2>(bufA, Wr[3], br[3], Wn[3], agg, bufB, src, dst,
                          stream);
    run_layer<128, 256, 2>(bufB, Wr[4], br[4], Wn[4], agg, bufA, src, dst,
                           stream);

    // global add pool over batch ids, then linear head
    const int nPool = N_GRAPHS * 256;
    zero_f32<<<(nPool + 255) / 256, 256, 0, stream>>>(pooled, nPool);
    const long pthr = (long)N_NODES * 64;
    pool_kernel<<<(int)((pthr + 255) / 256), 256, 0, stream>>>(bufA, batch,
                                                               pooled);
    final_linear<<<(N_GRAPHS * 2 + 255) / 256, 256, 0, stream>>>(pooled, Wlin,
                                                                 blin, out);
}